// SpanScorer_26276609917264
// MI455X (gfx1250) — compile-verified
//
#include <hip/hip_runtime.h>
#include <hip/hip_bf16.h>
#include <math.h>

typedef float v2f __attribute__((ext_vector_type(2)));
typedef float v8f __attribute__((ext_vector_type(8)));

#define L_TOK 4096
#define E_DIM 512
#define S_DIM 1024
#define H_RAW 150
#define HP    160          // 150 padded to 10 wmma tiles
#define T_ROWS 40915       // 10*L - 45
#define TP_ROWS 40960      // padded to multiple of 128 (gemm stripe)
#define MAX_WIN 10
#define KCHUNK 512         // K staged per LDS pass (32 KB slab)

// ---------------------------------------------------------------------------
// Zero-padding helpers: write EVERY element of the destination block, sourcing
// from src where in-range, else 0. Guarantees padded weight rows/cols are
// exactly zero so padded GEMM columns stay exactly zero through relu.
// ---------------------------------------------------------------------------
__global__ void pad2d_kernel(float* __restrict__ dst, const float* __restrict__ src,
                             int srcRowOff, int srcK, int srcN, int srcLd,
                             int Kp, int Nloc, int dstLd, int dstColBase)
{
    int idx = blockIdx.x * blockDim.x + threadIdx.x;
    int total = Kp * Nloc;
    if (idx >= total) return;
    int k = idx / Nloc;
    int n = idx - k * Nloc;
    float v = 0.f;
    if (k < srcK && n < srcN) v = src[(size_t)(srcRowOff + k) * srcLd + n];
    dst[(size_t)k * dstLd + dstColBase + n] = v;
}

__global__ void pad1d_kernel(float* __restrict__ dst, const float* __restrict__ src,
                             int srcN, int Np)
{
    int i = blockIdx.x * blockDim.x + threadIdx.x;
    if (i >= Np) return;
    dst[i] = (i < srcN) ? src[i] : 0.f;
}

// ---------------------------------------------------------------------------
// fp32 WMMA GEMM with LDS-staged weights:
//   C[stripe] = act(A[M x lda] @ W[K x ldw] + bias)
// Block = 128 threads (4 waves). Each block owns a 128-row M-stripe and one
// 16-col N-slab. The K x 16 weight slab is staged into LDS in KCHUNK chunks
// using CDNA5 async global->LDS copies (global_load_async_to_lds_b128,
// ASYNCcnt), then each wave computes TWO 16x16 m-tiles per k-step so every
// LDS B-fragment feeds 2 WMMAs.
//
// Async staging notes:
//  * The per-lane LDS byte address for the async op is derived from the real
//    generic pointer: on gfx12x the LDS-aperture flat address carries the LDS
//    byte offset in its low 32 bits (ISA 10.2 aperture mapping), so
//    (unsigned)(uintptr_t)&ldsWf[i] is the VDST value the async unit expects.
//  * The element pointer is ALSO passed as an asm operand so the LDS object's
//    address escapes -> LLVM must assume the asm writes it -> the later
//    ds_load B-fragment reads cannot be folded away (round-2 bug).
//
// Requirements (all guaranteed by the launcher): M % 128 == 0, K % 16 == 0,
// ldw % 4 == 0, all slabs 16B-aligned, grids exact (EXEC all-ones at WMMA).
//
// Fragment layout (ISA 7.12.2, 32-bit A 16x4):
//   A: lanes 0-15 hold A[m=lane][k..k+1]; lanes 16-31 hold A[m][k+2..k+3].
//   B: mirrored lane-half K split. C/D: vgpr v -> rows m0+v / m0+8+v.
// ---------------------------------------------------------------------------
__global__ __launch_bounds__(128)
void gemm_f32_wmma_lds(const float* __restrict__ A, const float* __restrict__ W,
                       const float* __restrict__ bias, float* __restrict__ C,
                       int K, int lda, int ldw, int ldc, int do_relu)
{
    __shared__ float ldsWf[KCHUNK * 16];   // 32 KB weight slab

    const int tid   = threadIdx.x;
    const int lane  = tid & 31;
    const int wave  = tid >> 5;
    const int n0    = blockIdx.y << 4;
    const int l15   = lane & 15;
    const int khalf = (lane >> 4) << 1;    // 0 for lanes 0-15, 2 for 16-31

    const int mbase = (blockIdx.x * 4 + wave) << 5;   // 2 m-tiles per wave
    const float* Arow0 = A + (size_t)(mbase + l15) * lda + khalf;
    const float* Arow1 = Arow0 + (size_t)16 * lda;

    v8f acc0 = {}, acc1 = {};

    for (int kbase = 0; kbase < K; kbase += KCHUNK) {
        const int kc = (K - kbase < KCHUNK) ? (K - kbase) : KCHUNK;

        // ---- async-stage the kc x 16 weight slab into LDS (16B quads) ----
        const int quads = kc << 2;                       // kc*16/4
        for (int q = tid; q < quads; q += 128) {
            const int row = q >> 2;
            const int qi  = q & 3;
            const float* gsrc = W + (size_t)(kbase + row) * ldw + n0 + (qi << 2);
            float* lptr = &ldsWf[q << 2];                // 4 floats per quad
            unsigned ldsoff = (unsigned)(uintptr_t)lptr; // low 32b = LDS byte addr
            asm volatile("global_load_async_to_lds_b128 %0, %1, off"
                         :: "v"(ldsoff), "v"(gsrc), "v"(lptr)   // lptr: escape LDS obj
                         : "memory");
        }
#if __has_builtin(__builtin_amdgcn_s_wait_asynccnt)
        __builtin_amdgcn_s_wait_asynccnt(0);
#else
        asm volatile("s_wait_asynccnt 0x0" ::: "memory");
#endif
        __syncthreads();

        // ---- accumulate: 2 WMMAs per LDS B-fragment ----------------------
        for (int k = 0; k < kc; k += 16) {
            __builtin_prefetch(Arow0 + kbase + k + 256, 0, 1);  // global_prefetch_b8
#pragma unroll
            for (int kk = 0; kk < 16; kk += 4) {
                const int kr = k + kk;
                v2f b;
                b.x = ldsWf[(kr + khalf) * 16 + l15];
                b.y = ldsWf[(kr + khalf + 1) * 16 + l15];
                v2f a0 = *(const v2f*)(Arow0 + kbase + kr);
                v2f a1 = *(const v2f*)(Arow1 + kbase + kr);
                acc0 = __builtin_amdgcn_wmma_f32_16x16x4_f32(
                           false, a0, false, b, (short)0, acc0, false, false);
                acc1 = __builtin_amdgcn_wmma_f32_16x16x4_f32(
                           false, a1, false, b, (short)0, acc1, false, false);
            }
        }
        __syncthreads();   // slab reusable next chunk
    }

    // ---- epilogue --------------------------------------------------------
    const int crow = mbase + ((lane >> 4) << 3);
    const int ccol = n0 + l15;
    const float bv = bias[ccol];
#pragma unroll
    for (int v = 0; v < 8; ++v) {
        float x0 = acc0[v] + bv;
        float x1 = acc1[v] + bv;
        C[(size_t)(crow + v) * ldc + ccol]      = do_relu ? fmaxf(x0, 0.f) : x0;
        C[(size_t)(crow + 16 + v) * ldc + ccol] = do_relu ? fmaxf(x1, 0.f) : x1;
    }
}

// ---------------------------------------------------------------------------
// Combine: for each window row t=(n,s): windowed softmax over attn logits,
// h1 = relu(Ua[s] + Ub[s+n-1] + sum_i w_i * P[s+i] + b1)   -> H1[t][0:160]
// One wave per row (n uniform per wave -> no divergence in the hot loop).
// ---------------------------------------------------------------------------
__global__ __launch_bounds__(256)
void combine_kernel(const float* __restrict__ U,    // [L x 320]  (Ua | Ub)
                    const float* __restrict__ P,    // [L x 160]
                    const float* __restrict__ ATT,  // [L x 16], col 0 = logits
                    const float* __restrict__ b1p,  // [160]
                    float* __restrict__ H1)         // [TP_ROWS x 160]
{
    int row = blockIdx.x * blockDim.y + threadIdx.y;
    if (row >= TP_ROWS) return;
    int lane = threadIdx.x;               // 0..31

    if (row >= T_ROWS) {                  // zero-fill padded rows
#pragma unroll
        for (int q = 0; q < 5; ++q)
            H1[(size_t)row * HP + lane + 32 * q] = 0.f;
        return;
    }

    // map flat row -> (window size n, start s); blocks for n are contiguous
    int n = 1, s = row, off = 0;
#pragma unroll
    for (int k = 1; k <= MAX_WIN; ++k) {
        int len = L_TOK - k + 1;
        if (row < off + len) { n = k; s = row - off; break; }
        off += len;
    }

    // masked 10-wide softmax over attn logits (fixed trip count, no spills)
    float w[MAX_WIN];
    float mx = -INFINITY;
#pragma unroll
    for (int i = 0; i < MAX_WIN; ++i) {
        int idx = min(s + i, L_TOK - 1);
        float a = (i < n) ? ATT[(size_t)idx * 16] : -INFINITY;
        w[i] = a;
        mx = fmaxf(mx, a);
    }
    float sum = 0.f;
#pragma unroll
    for (int i = 0; i < MAX_WIN; ++i) { w[i] = __expf(w[i] - mx); sum += w[i]; }
    float inv = 1.f / sum;

    int e = s + n - 1;
#pragma unroll
    for (int q = 0; q < 5; ++q) {
        int j = lane + 32 * q;
        float h = U[(size_t)s * 320 + j] + U[(size_t)e * 320 + 160 + j] + b1p[j];
#pragma unroll
        for (int i = 0; i < MAX_WIN; ++i) {
            int idx = min(s + i, L_TOK - 1);
            h += (w[i] * inv) * P[(size_t)idx * HP + j];
        }
        H1[(size_t)row * HP + j] = fmaxf(h, 0.f);
    }
}

__global__ void extract_kernel(const float* __restrict__ S3, float* __restrict__ out)
{
    int t = blockIdx.x * blockDim.x + threadIdx.x;
    if (t < T_ROWS) out[t] = S3[(size_t)t * 16];
}

// ---------------------------------------------------------------------------
extern "C" void kernel_launch(void* const* d_in, const int* in_sizes, int n_in,
                              void* d_out, int out_size, void* d_ws, size_t ws_size,
                              hipStream_t stream)
{
    const float* embeds = (const float*)d_in[0];
    const float* states = (const float*)d_in[1];
    const float* aW1 = (const float*)d_in[2];
    const float* ab1 = (const float*)d_in[3];
    const float* aW2 = (const float*)d_in[4];
    const float* ab2 = (const float*)d_in[5];
    const float* aW3 = (const float*)d_in[6];
    const float* ab3 = (const float*)d_in[7];
    const float* sW1 = (const float*)d_in[8];
    const float* sb1 = (const float*)d_in[9];
    const float* sW2 = (const float*)d_in[10];
    const float* sb2 = (const float*)d_in[11];
    const float* sW3 = (const float*)d_in[12];
    const float* sb3 = (const float*)d_in[13];
    float* out = (float*)d_out;

    // workspace carve-up (floats); all slab bases are multiples of 16 floats
    float* ws = (float*)d_ws;
    size_t off = 0;
    auto take = [&](size_t n) { float* p = ws + off; off += n; return p; };
    float* aW1p = take((size_t)S_DIM * HP);       // 1024x160
    float* aW2p = take((size_t)HP * HP);          // 160x160
    float* aW3p = take((size_t)HP * 16);          // 160x16
    float* WabP = take((size_t)S_DIM * 2 * HP);   // 1024x320  [Wa | Wb]
    float* WcP  = take((size_t)E_DIM * HP);       // 512x160
    float* sW2p = take((size_t)HP * HP);          // 160x160
    float* sW3p = take((size_t)HP * 16);          // 160x16
    float* ab1p = take(HP);
    float* ab2p = take(HP);
    float* ab3p = take(16);
    float* sb1p = take(HP);
    float* sb2p = take(HP);
    float* sb3p = take(16);
    float* zb   = take(2 * HP);                   // zero bias [320]
    float* A1   = take((size_t)L_TOK * HP);       // 4096x160
    float* A2   = take((size_t)L_TOK * HP);       // 4096x160
    float* ATT  = take((size_t)L_TOK * 16);       // 4096x16
    float* U    = take((size_t)L_TOK * 2 * HP);   // 4096x320
    float* P    = take((size_t)L_TOK * HP);       // 4096x160
    float* H1   = take((size_t)TP_ROWS * HP);     // 40960x160
    float* H2   = take((size_t)TP_ROWS * HP);     // 40960x160
    float* S3   = take((size_t)TP_ROWS * 16);     // 40960x16
    (void)ws_size; (void)in_sizes; (void)n_in; (void)out_size;

    const int PT = 256;
    auto pgrid = [](size_t total) { return (unsigned)((total + 255) / 256); };

    // ---- stage 0: pad weights/biases into zero-filled tiles -------------
    pad2d_kernel<<<pgrid((size_t)S_DIM * HP), PT, 0, stream>>>(aW1p, aW1, 0, S_DIM, H_RAW, H_RAW, S_DIM, HP, HP, 0);
    pad2d_kernel<<<pgrid((size_t)HP * HP),    PT, 0, stream>>>(aW2p, aW2, 0, H_RAW, H_RAW, H_RAW, HP, HP, HP, 0);
    pad2d_kernel<<<pgrid((size_t)HP * 16),    PT, 0, stream>>>(aW3p, aW3, 0, H_RAW, 1, 1, HP, 16, 16, 0);
    pad2d_kernel<<<pgrid((size_t)S_DIM * HP), PT, 0, stream>>>(WabP, sW1, 0,     S_DIM, H_RAW, H_RAW, S_DIM, HP, 2 * HP, 0);
    pad2d_kernel<<<pgrid((size_t)S_DIM * HP), PT, 0, stream>>>(WabP, sW1, S_DIM, S_DIM, H_RAW, H_RAW, S_DIM, HP, 2 * HP, HP);
    pad2d_kernel<<<pgrid((size_t)E_DIM * HP), PT, 0, stream>>>(WcP,  sW1, 2 * S_DIM, E_DIM, H_RAW, H_RAW, E_DIM, HP, HP, 0);
    pad2d_kernel<<<pgrid((size_t)HP * HP),    PT, 0, stream>>>(sW2p, sW2, 0, H_RAW, H_RAW, H_RAW, HP, HP, HP, 0);
    pad2d_kernel<<<pgrid((size_t)HP * 16),    PT, 0, stream>>>(sW3p, sW3, 0, H_RAW, 1, 1, HP, 16, 16, 0);
    pad1d_kernel<<<1, PT, 0, stream>>>(ab1p, ab1, H_RAW, HP);
    pad1d_kernel<<<1, PT, 0, stream>>>(ab2p, ab2, H_RAW, HP);
    pad1d_kernel<<<1, PT, 0, stream>>>(ab3p, ab3, 1, 16);
    pad1d_kernel<<<1, PT, 0, stream>>>(sb1p, sb1, H_RAW, HP);
    pad1d_kernel<<<1, PT, 0, stream>>>(sb2p, sb2, H_RAW, HP);
    pad1d_kernel<<<1, PT, 0, stream>>>(sb3p, sb3, 1, 16);
    pad1d_kernel<<<2, PT, 0, stream>>>(zb, ab1, 0, 2 * HP);   // zeros (src unread)

    // ---- stage 1: attn scorer + layer-1 precompute (WMMA GEMMs) ---------
    const unsigned MG_L  = L_TOK / 128;     // 32 stripes of 128 rows
    const unsigned MG_T  = TP_ROWS / 128;   // 320 stripes
    // A1 = relu(states @ aW1p + ab1)          [4096 x 160], K=1024
    gemm_f32_wmma_lds<<<dim3(MG_L, HP / 16), 128, 0, stream>>>(states, aW1p, ab1p, A1, S_DIM, S_DIM, HP, HP, 1);
    // A2 = relu(A1 @ aW2p + ab2)              [4096 x 160], K=160
    gemm_f32_wmma_lds<<<dim3(MG_L, HP / 16), 128, 0, stream>>>(A1, aW2p, ab2p, A2, HP, HP, HP, HP, 1);
    // ATT = A2 @ aW3p + ab3                   [4096 x 16],  K=160 (col 0 = logits)
    gemm_f32_wmma_lds<<<dim3(MG_L, 1), 128, 0, stream>>>(A2, aW3p, ab3p, ATT, HP, HP, 16, 16, 0);
    // U = states @ [Wa|Wb]                    [4096 x 320], K=1024
    gemm_f32_wmma_lds<<<dim3(MG_L, 2 * HP / 16), 128, 0, stream>>>(states, WabP, zb, U, S_DIM, S_DIM, 2 * HP, 2 * HP, 0);
    // P = embeds @ Wc                         [4096 x 160], K=512
    gemm_f32_wmma_lds<<<dim3(MG_L, HP / 16), 128, 0, stream>>>(embeds, WcP, zb, P, E_DIM, E_DIM, HP, HP, 0);

    // ---- stage 2: windowed softmax + layer-1 assembly -------------------
    combine_kernel<<<TP_ROWS / 8, dim3(32, 8), 0, stream>>>(U, P, ATT, sb1p, H1);

    // ---- stage 3: layers 2-3 of the big scorer (WMMA GEMMs) -------------
    gemm_f32_wmma_lds<<<dim3(MG_T, HP / 16), 128, 0, stream>>>(H1, sW2p, sb2p, H2, HP, HP, HP, HP, 1);
    gemm_f32_wmma_lds<<<dim3(MG_T, 1), 128, 0, stream>>>(H2, sW3p, sb3p, S3, HP, HP, 16, 16, 0);

    // ---- stage 4: extract column 0 --------------------------------------
    extract_kernel<<<(T_ROWS + 255) / 256, 256, 0, stream>>>(S3, out);
}